// ProposalCaffe_5970004541863
// MI455X (gfx1250) — compile-verified
//
#include <hip/hip_runtime.h>
#include <cmath>
#include <cstdint>
#include <cstddef>

#ifndef __has_builtin
#define __has_builtin(x) 0
#endif

#if __has_builtin(__builtin_amdgcn_global_load_async_to_lds_b128)
#define HAVE_ASYNC_B128 1
#else
#define HAVE_ASYNC_B128 0
#endif

typedef float v2f __attribute__((ext_vector_type(2)));
typedef float v8f __attribute__((ext_vector_type(8)));
typedef int   v4i __attribute__((vector_size(4 * sizeof(int))));

#define NA        36864      // 64*64*9 anchors
#define NSORT     65536      // padded to pow2 for bitonic
#define NTOP      6000       // PRE_NMS_TOPN
#define NOUT      300        // POST_NMS_TOPN
#define NWORDS    188        // ceil(6000/32)
#define IMSZ      1024.0f    // 64 * 16
#define NMS_TH    0.5f

struct AnchorTab { float a[9][4]; };

// ---------------------------------------------------------------- decode ----
__global__ void decode_kernel(const float* __restrict__ scores,
                              const float* __restrict__ deltas,
                              float4* __restrict__ prop,
                              unsigned long long* __restrict__ keys,
                              AnchorTab at) {
  int i = blockIdx.x * 256 + threadIdx.x;            // 0..65535
  if (i >= NA) { keys[i] = ~0ull; return; }          // padding sinks to end
  int a    = i % 9;
  int cell = i / 9;
  int wx = cell & 63;
  int hy = cell >> 6;
  float sx = wx * 16.0f, sy = hy * 16.0f;
  float x1a = at.a[a][0] + sx, y1a = at.a[a][1] + sy;
  float x2a = at.a[a][2] + sx, y2a = at.a[a][3] + sy;
  float aw  = x2a - x1a + 1.0f, ah = y2a - y1a + 1.0f;
  float acx = x1a + 0.5f * aw,  acy = y1a + 0.5f * ah;
  const float* dp = deltas + (size_t)cell * 36 + a * 4;
  float dx = dp[0], dy = dp[1], dw = dp[2], dh = dp[3];
  float score = scores[(size_t)cell * 18 + 9 + a];   // fg channel A..2A-1
  float pcx = dx * aw + acx, pcy = dy * ah + acy;
  float pw  = expf(dw) * aw, ph  = expf(dh) * ah;
  float x1 = fminf(fmaxf(pcx - 0.5f * pw, 0.0f), IMSZ - 1.0f);
  float y1 = fminf(fmaxf(pcy - 0.5f * ph, 0.0f), IMSZ - 1.0f);
  float x2 = fminf(fmaxf(pcx + 0.5f * pw, 0.0f), IMSZ - 1.0f);
  float y2 = fminf(fmaxf(pcy + 0.5f * ph, 0.0f), IMSZ - 1.0f);
  prop[i] = make_float4(x1, y1, x2, y2);
  bool valid = ((x2 - x1 + 1.0f) >= 16.0f) && ((y2 - y1 + 1.0f) >= 16.0f);
  // positive-float bit pattern is order preserving; +1 distinguishes 0.0 from -inf
  unsigned skey = valid ? (__float_as_uint(score) + 1u) : 0u;
  keys[i] = ((unsigned long long)(~skey) << 32) | (unsigned)i; // asc == score desc, stable
}

// ---------------------------------------------------- bitonic sort (u64) ----
__global__ void bitonic_global(unsigned long long* keys, int k, int j) {
  int i = blockIdx.x * 256 + threadIdx.x;
  int l = i ^ j;
  if (l > i) {
    unsigned long long a = keys[i], b = keys[l];
    bool up = ((i & k) == 0);
    if ((a > b) == up) { keys[i] = b; keys[l] = a; }
  }
}

__global__ void bitonic_fused(unsigned long long* keys, int k) {
  __shared__ unsigned long long tile[512];
  int base = blockIdx.x * 512;
  int tid  = threadIdx.x;
  tile[tid]       = keys[base + tid];
  tile[tid + 256] = keys[base + tid + 256];
  __syncthreads();
  int jstart = (k >> 1) < 256 ? (k >> 1) : 256;
  for (int j = jstart; j > 0; j >>= 1) {
    int i = ((tid & ~(j - 1)) << 1) | (tid & (j - 1));
    int l = i | j;
    unsigned long long a = tile[i], b = tile[l];
    bool up = (((base + i) & k) == 0);
    if ((a > b) == up) { tile[i] = b; tile[l] = a; }
    __syncthreads();
  }
  keys[base + tid]       = tile[tid];
  keys[base + tid + 256] = tile[tid + 256];
}

// ---------------------------------------------------------------- gather ----
__global__ void gather_top(const unsigned long long* __restrict__ keys,
                           const float4* __restrict__ prop,
                           float4* __restrict__ topbox,
                           unsigned* __restrict__ topvalid) {
  int i = blockIdx.x * 256 + threadIdx.x;
  if (i >= NTOP) return;
  unsigned long long k = keys[i];
  unsigned idx  = (unsigned)(k & 0xFFFFFFFFull);
  unsigned skey = ~(unsigned)(k >> 32);
  idx = idx < (unsigned)NA ? idx : 0u;
  topbox[i]   = prop[idx];   // invalid (min-size) boxes still participate as suppressors
  topvalid[i] = skey;        // 0 <=> score was -inf (fails isfinite at the end)
}

// ----------------------------------------------------- NMS bitmask build ----
__global__ void nms_mask(const float4* __restrict__ topbox,
                         unsigned* __restrict__ mask) {
  __shared__ float4 scol[32];
  int colw = blockIdx.x;                          // 0..187
  int col0 = colw * 32;
  int row  = blockIdx.y * 128 + threadIdx.x;
  if (threadIdx.x < 32) {
#if HAVE_ASYNC_B128
    __builtin_amdgcn_global_load_async_to_lds_b128(
        (__attribute__((address_space(1))) v4i*)(topbox + col0 + (int)threadIdx.x),
        (__attribute__((address_space(3))) v4i*)&scol[threadIdx.x], 0, 0);
#else
    int j = col0 + (int)threadIdx.x;
    scol[threadIdx.x] = topbox[j < NTOP ? j : NTOP - 1];
#endif
  }
#if HAVE_ASYNC_B128
#if __has_builtin(__builtin_amdgcn_s_wait_asynccnt)
  __builtin_amdgcn_s_wait_asynccnt(0);
#else
  asm volatile("s_wait_asynccnt 0" ::: "memory");
#endif
#endif
  __syncthreads();
  if (row >= NTOP) return;
  float4 rb = topbox[row];
  float rarea = (rb.z - rb.x + 1.0f) * (rb.w - rb.y + 1.0f);
  unsigned bits = 0u;
  #pragma unroll 4
  for (int c = 0; c < 32; ++c) {
    int j = col0 + c;
    if (j > row && j < NTOP) {
      float4 cb = scol[c];
      float carea = (cb.z - cb.x + 1.0f) * (cb.w - cb.y + 1.0f);
      float xx1 = fmaxf(rb.x, cb.x), yy1 = fmaxf(rb.y, cb.y);
      float xx2 = fminf(rb.z, cb.z), yy2 = fminf(rb.w, cb.w);
      float iw = fmaxf(xx2 - xx1 + 1.0f, 0.0f);
      float ih = fmaxf(yy2 - yy1 + 1.0f, 0.0f);
      float inter = iw * ih;
      float iou = inter / (rarea + carea - inter);
      if (iou > NMS_TH) bits |= (1u << c);
    }
  }
  mask[(size_t)row * NWORDS + colw] = bits;
}

// ------------------------------------------------- serial greedy NMS scan ---
__global__ void nms_scan(const unsigned* __restrict__ mask,
                         const unsigned* __restrict__ topvalid,
                         unsigned* __restrict__ keep) {
  __shared__ int sAlive;
  int tid = threadIdx.x;
  unsigned rem = 0u;                       // each lane owns removed-word `tid`
  for (int i = 0; i < NTOP; ++i) {
    if (tid == (i >> 5)) {
      int alive = ((rem >> (i & 31)) & 1u) ? 0 : 1;
      sAlive = alive;
      keep[i] = (alive && topvalid[i] != 0u) ? 1u : 0u;  // keep &= isfinite
    }
    __syncthreads();
    if (sAlive && tid < NWORDS) rem |= mask[(size_t)i * NWORDS + tid];
    if (tid < NWORDS && (i + 1) < NTOP)
      __builtin_prefetch(&mask[(size_t)(i + 1) * NWORDS + tid], 0, 0); // global_prefetch_b8
    __syncthreads();
  }
}

// ------------------------- compaction via WMMA tensor-core block prefix-sum -
__global__ void compact_out(const unsigned* __restrict__ keep,
                            const float4* __restrict__ topbox,
                            float* __restrict__ out) {
  __shared__ float sval[256];
  __shared__ float sscan[256];
  __shared__ float srowoff[16];
  __shared__ float sbase;
  __shared__ float stot;
  int tid = threadIdx.x;
  for (int i = tid; i < NOUT * 4; i += 256) out[i] = 0.0f;   // rows beyond kept -> 0
  if (tid == 0) sbase = 0.0f;
  __syncthreads();

  for (int chunk = 0; chunk < 24; ++chunk) {                 // 24*256 >= 6000
    int i = chunk * 256 + tid;
    float f = (i < NTOP && keep[i]) ? 1.0f : 0.0f;
    sval[tid] = f;
    __syncthreads();

    // Wave 0: inclusive row-prefix of V(16x16) = V x U (upper-triangular ones),
    // as 4 chained v_wmma_f32_16x16x4_f32 accumulations. EXEC all-1 in wave 0.
    if (tid < 32) {
      int L = tid;
      int m = L & 15;        // A-matrix row / C-matrix column group
      int g = L >> 4;        // lane half selects K pair
      v8f c = {};
      #pragma unroll
      for (int blk = 0; blk < 4; ++blk) {
        int k0 = 4 * blk + 2 * g;
        v2f a, b;
        a.x = sval[m * 16 + k0];
        a.y = sval[m * 16 + k0 + 1];
        b.x = (k0     <= m) ? 1.0f : 0.0f;   // U[k][n], n == L&15
        b.y = (k0 + 1 <= m) ? 1.0f : 0.0f;
        c = __builtin_amdgcn_wmma_f32_16x16x4_f32(
                false, a, false, b, (short)0, c, false, false);
      }
      #pragma unroll
      for (int v = 0; v < 8; ++v) {
        int M = v + 8 * g;                   // C/D layout: lanes16-31 hold M+8
        sscan[M * 16 + m] = c[v];
      }
    }
    __syncthreads();

    if (tid == 0) {                          // tiny 16-wide exclusive scan of row sums
      float acc = 0.0f;
      for (int r = 0; r < 16; ++r) { srowoff[r] = acc; acc += sscan[r * 16 + 15]; }
      stot = acc;
    }
    __syncthreads();

    float incl = sscan[tid] + srowoff[tid >> 4];   // inclusive rank within chunk
    if (f > 0.5f) {
      int rank = (int)(sbase + incl) - 1;
      if (rank < NOUT) {
        float4 b = topbox[i];
        const float inv = 1.0f / IMSZ;
        out[rank * 4 + 0] = b.x * inv;
        out[rank * 4 + 1] = b.y * inv;
        out[rank * 4 + 2] = b.z * inv;
        out[rank * 4 + 3] = b.w * inv;
      }
    }
    __syncthreads();
    if (tid == 0) sbase += stot;
    __syncthreads();
  }
}

// ------------------------------------------------------------------ host ----
extern "C" void kernel_launch(void* const* d_in, const int* in_sizes, int n_in,
                              void* d_out, int out_size, void* d_ws, size_t ws_size,
                              hipStream_t stream) {
  (void)in_sizes; (void)n_in; (void)out_size; (void)ws_size;
  const float* scores = (const float*)d_in[0];   // (1,64,64,18) f32
  const float* deltas = (const float*)d_in[1];   // (1,64,64,36) f32
  float* out = (float*)d_out;                    // (300,4) f32

  char* ws = (char*)d_ws;
  float4*             prop     = (float4*)(ws);                       // 36864*16 = 589824
  unsigned long long* keys     = (unsigned long long*)(ws + 589824);  // 65536*8  = 524288
  float4*             topbox   = (float4*)(ws + 1114112);             // 6016*16  = 96256
  unsigned*           topvalid = (unsigned*)(ws + 1210368);           // 6016*4   = 24064
  unsigned*           mask     = (unsigned*)(ws + 1234432);           // 6000*188*4 = 4512000
  unsigned*           keep     = (unsigned*)(ws + 5746432);           // 6016*4

  // Anchors computed exactly like _generate_anchors (np.round == half-to-even)
  AnchorTab at;
  {
    const double ratios[3] = {0.5, 1.0, 2.0};
    const double scales[3] = {8.0, 16.0, 32.0};
    int t = 0;
    for (int r = 0; r < 3; ++r) {
      double wsz = std::nearbyint(std::sqrt(256.0 / ratios[r]));
      double hsz = std::nearbyint(wsz * ratios[r]);
      for (int s = 0; s < 3; ++s) {
        double W = wsz * scales[s], H = hsz * scales[s];
        at.a[t][0] = (float)(7.5 - (W - 1.0) * 0.5);
        at.a[t][1] = (float)(7.5 - (H - 1.0) * 0.5);
        at.a[t][2] = (float)(7.5 + (W - 1.0) * 0.5);
        at.a[t][3] = (float)(7.5 + (H - 1.0) * 0.5);
        ++t;
      }
    }
  }

  decode_kernel<<<NSORT / 256, 256, 0, stream>>>(scores, deltas, prop, keys, at);

  // full bitonic sort of 65536 u64 keys (ascending == score descending, stable)
  for (int k = 2; k <= NSORT; k <<= 1) {
    for (int j = k >> 1; j >= 512; j >>= 1)
      bitonic_global<<<NSORT / 256, 256, 0, stream>>>(keys, k, j);
    bitonic_fused<<<NSORT / 512, 256, 0, stream>>>(keys, k);
  }

  gather_top<<<24, 256, 0, stream>>>(keys, prop, topbox, topvalid);
  nms_mask<<<dim3(NWORDS, (NTOP + 127) / 128), 128, 0, stream>>>(topbox, mask);
  nms_scan<<<1, 256, 0, stream>>>(mask, topvalid, keep);
  compact_out<<<1, 256, 0, stream>>>(keep, topbox, out);
}